// HyperbolicKuramoto_10213432230576
// MI455X (gfx1250) — compile-verified
//
#include <hip/hip_runtime.h>
#include <math.h>

typedef __attribute__((ext_vector_type(2))) float v2f;
typedef __attribute__((ext_vector_type(8))) float v8f;

#define NN 4096
#define NSTEPS 100
#define INV2N (1.0f / 8192.0f)   // 1/(2N)

// ---------------------------------------------------------------------------
// Matvec: U = K @ x, V = K @ y via V_WMMA_F32_16X16X4_F32.
// One block (8 wave32) per 32-row stripe of K (two 16-row WMMA tiles sharing
// the B operand). Each wave covers a 512-deep k slice; partial C tiles are
// reduced through LDS. B tile: col 0 = x, col 1 = y, cols 2..15 = 0, built
// branch-free with per-lane 0/1 mask multiplies (keeps EXEC all-ones).
// Software-pipelined: chunk c+1's loads are in flight during chunk c's math,
// so the scheduler can wait on loadcnt<=3 instead of draining to 0.
// ---------------------------------------------------------------------------
__global__ __launch_bounds__(256) void hk_matvec(
    const float* __restrict__ K, const float2* __restrict__ z,
    float* __restrict__ U, float* __restrict__ V) {
  const int lane = threadIdx.x & 31;
  const int wave = threadIdx.x >> 5;        // 0..7
  const int n    = lane & 15;               // B/C column; A row (M) in tile
  const int half = lane >> 4;               // 0: k0,k0+1 ; 1: k0+2,k0+3
  const int base = blockIdx.x * 32;         // first row of this block's stripe

  const float* __restrict__ Krow0 = K + (size_t)(base + n) * NN;
  const float* __restrict__ Krow1 = K + (size_t)(base + 16 + n) * NN;

  const float mx = (n == 0) ? 1.0f : 0.0f;  // select x into column 0
  const float my = (n == 1) ? 1.0f : 0.0f;  // select y into column 1

  v8f accA = {};
  v8f accB = {};

  const int CH = (NN / 8) / 4;              // 128 chunks of k-depth 4 per wave
  int kk = wave * (NN / 8) + 2 * half;

  // Pipeline prologue: first chunk's loads.
  v2f   aA = *(const v2f*)(Krow0 + kk);
  v2f   aB = *(const v2f*)(Krow1 + kk);
  float4 zz = *(const float4*)(z + kk);

  #pragma unroll 4
  for (int c = 0; c < CH - 1; ++c) {
    const int kn = kk + 4;
    // Prefetch chunk c+1 (always in-bounds: c+1 <= CH-1).
    v2f   aA2 = *(const v2f*)(Krow0 + kn);
    v2f   aB2 = *(const v2f*)(Krow1 + kn);
    float4 zz2 = *(const float4*)(z + kn);
    // Compute chunk c.
    v2f b;
    b.x = fmaf(zz.y, my, zz.x * mx);
    b.y = fmaf(zz.w, my, zz.z * mx);
    accA = __builtin_amdgcn_wmma_f32_16x16x4_f32(
        false, aA, false, b, (short)0, accA, false, false);
    accB = __builtin_amdgcn_wmma_f32_16x16x4_f32(
        false, aB, false, b, (short)0, accB, false, false);
    aA = aA2; aB = aB2; zz = zz2;
    kk = kn;
  }
  // Epilogue: last chunk (no prefetch -> no out-of-bounds reads).
  {
    v2f b;
    b.x = fmaf(zz.y, my, zz.x * mx);
    b.y = fmaf(zz.w, my, zz.z * mx);
    accA = __builtin_amdgcn_wmma_f32_16x16x4_f32(
        false, aA, false, b, (short)0, accA, false, false);
    accB = __builtin_amdgcn_wmma_f32_16x16x4_f32(
        false, aB, false, b, (short)0, accB, false, false);
  }

  // Cross-wave reduction of the 8 partial C tiles for each of the two tiles.
  __shared__ float red[2][8][32][8];        // 16 KB
  #pragma unroll
  for (int j = 0; j < 8; ++j) {
    red[0][wave][lane][j] = accA[j];
    red[1][wave][lane][j] = accB[j];
  }
  __syncthreads();

  if (wave < 2) {                           // wave 0 -> tile A, wave 1 -> tile B
    const int t = wave;
    float s[8];
    #pragma unroll
    for (int j = 0; j < 8; ++j) s[j] = 0.0f;
    #pragma unroll
    for (int w = 0; w < 8; ++w)
      #pragma unroll
      for (int j = 0; j < 8; ++j) s[j] += red[t][w][lane][j];
    // C layout: VGPR j, lane L -> M = j + 8*(L>>4), N = L&15.
    const int M0 = base + t * 16 + 8 * half;
    if (n == 0) {
      #pragma unroll
      for (int j = 0; j < 8; ++j) U[M0 + j] = s[j];
    } else if (n == 1) {
      #pragma unroll
      for (int j = 0; j < 8; ++j) V[M0 + j] = s[j];
    }
  }
}

// ---------------------------------------------------------------------------
// Pointwise step:  S = u - i v,  T = u + i v  (K real)
//   dz = -inv2n * S * z^2 + omega * z + inv2n * T ;  z += dt*dz ; clamp |z|.
// ---------------------------------------------------------------------------
__global__ __launch_bounds__(256) void hk_update(
    const float2* __restrict__ zin, const float* __restrict__ U,
    const float* __restrict__ V, float2* __restrict__ zout,
    const float* __restrict__ omega_p, const float* __restrict__ dt_p) {
  const int i = blockIdx.x * blockDim.x + threadIdx.x;
  if (i >= NN) return;
  const float omega = omega_p[0];
  const float dt = dt_p[0];

  float2 zi = zin[i];
  float x = zi.x, y = zi.y;
  float u = U[i], v = V[i];

  float a = x * x - y * y;       // Re(z^2)
  float b = 2.0f * x * y;        // Im(z^2)
  // S*z^2 = (u - iv)(a + ib) = (ua + vb) + i(ub - va)
  float dre = -INV2N * (u * a + v * b) + omega * x + INV2N * u;
  float dim = -INV2N * (u * b - v * a) + omega * y + INV2N * v;
  x += dt * dre;
  y += dt * dim;

  float r = sqrtf(x * x + y * y);
  if (r >= 0.999f) {
    float sc = 0.999f / r;
    x *= sc;
    y *= sc;
  }
  zout[i] = make_float2(x, y);
}

// Final clamp on the (N,2) output (mirrors the reference's post-scan clamp).
__global__ __launch_bounds__(256) void hk_finalclamp(float2* __restrict__ z) {
  const int i = blockIdx.x * blockDim.x + threadIdx.x;
  if (i >= NN) return;
  float2 zi = z[i];
  float r = sqrtf(zi.x * zi.x + zi.y * zi.y);
  if (r >= 0.999f) {
    float sc = 0.999f / r;
    z[i] = make_float2(zi.x * sc, zi.y * sc);
  }
}

extern "C" void kernel_launch(void* const* d_in, const int* in_sizes, int n_in,
                              void* d_out, int out_size, void* d_ws, size_t ws_size,
                              hipStream_t stream) {
  const float*  Zin   = (const float*)d_in[0];   // (N,2) fp32, interleaved re/im
  const float*  K     = (const float*)d_in[1];   // (N,N) fp32 row-major
  const float*  omega = (const float*)d_in[2];   // scalar
  const float*  dt    = (const float*)d_in[3];   // scalar
  // d_in[4] = steps (Python int, == NSTEPS; compile-time constant here)

  char* ws = (char*)d_ws;
  float2* bufA = (float2*)ws;                                    // 32 KB
  float2* bufB = (float2*)(ws + (size_t)NN * sizeof(float2));    // 32 KB
  float*  U    = (float*)(ws + (size_t)2 * NN * sizeof(float2)); // 16 KB
  float*  V    = U + NN;                                         // 16 KB

  const float2* cur = (const float2*)Zin;
  for (int s = 0; s < NSTEPS; ++s) {
    float2* next = (s == NSTEPS - 1) ? (float2*)d_out
                                     : ((s & 1) ? bufB : bufA);
    hk_matvec<<<NN / 32, 256, 0, stream>>>(K, cur, U, V);
    hk_update<<<NN / 256, 256, 0, stream>>>(cur, U, V, next, omega, dt);
    cur = next;
  }
  hk_finalclamp<<<NN / 256, 256, 0, stream>>>((float2*)d_out);
}